// RoIAlign_15719580304193
// MI455X (gfx1250) — compile-verified
//
#include <hip/hip_runtime.h>
#include <hip/hip_bf16.h>
#include <stdint.h>

// ---------------- problem constants (from reference) ----------------
#define OHP 7
#define OWP 7
#define NROIS 2048
#define FB 4
#define FC 256
#define FH 200
#define FW 336
#define FHW (FH * FW)            // 67200 (divisible by 32)
#define SPATIAL_SCALE 0.25f
#define NBINS (OHP * OWP)        // 49
#define ROI_ELEMS (FC * NBINS)   // 12544 floats per roi

// ---------------- CDNA5 data-mover paths (guarded) ------------------
#if defined(__has_builtin)
#if __has_builtin(__builtin_amdgcn_tensor_load_to_lds) && \
    __has_builtin(__builtin_amdgcn_s_wait_tensorcnt)
#define USE_TDM 1
#endif
#if __has_builtin(__builtin_amdgcn_global_load_async_to_lds_b32) && \
    __has_builtin(__builtin_amdgcn_s_wait_asynccnt)
#define USE_ASYNC_LDS 1
#endif
#if __has_builtin(__builtin_amdgcn_global_store_async_from_lds_b32) && \
    __has_builtin(__builtin_amdgcn_s_wait_asynccnt)
#define USE_ASYNC_ST 1
#endif
#endif
#ifndef USE_TDM
#define USE_TDM 0
#endif
#ifndef USE_ASYNC_LDS
#define USE_ASYNC_LDS 0
#endif
#ifndef USE_ASYNC_ST
#define USE_ASYNC_ST 0
#endif

typedef __attribute__((address_space(1))) int* as1i_t;
typedef __attribute__((address_space(3))) int* as3i_t;
typedef unsigned int v4u __attribute__((ext_vector_type(4)));
typedef int          v8i __attribute__((ext_vector_type(8)));
typedef int          v4i __attribute__((ext_vector_type(4)));

// =====================================================================
// Kernel 1: NCHW -> NHWC transpose, 32x32 tiles staged through LDS.
// Preferred path: one TENSOR_LOAD_TO_LDS (TDM) per tile — 2D descriptor
// with tensor_dim0_stride = FHW and hardware row padding (pad_interval =
// 32 dwords, pad_amount = 1 dword) producing the bank-conflict-free
// stride-33 LDS layout directly.  Fallbacks: per-lane async-to-LDS, then
// plain float4 + ds stores.  Store phase: coalesced float4 along C.
// =====================================================================
__global__ void nchw_to_nhwc(const float* __restrict__ src,
                             float* __restrict__ dst) {
    __shared__ float tile[32][33];

    const int hw0 = blockIdx.x * 32;   // 0..67199
    const int c0  = blockIdx.y * 32;   // 0..255
    const int b   = blockIdx.z;        // 0..3
    const int t   = threadIdx.x;       // 0..255

#if USE_TDM
    if (t < 32) {   // wave 0 issues the DMA (descriptor is block-uniform)
        const uint64_t ga = (uint64_t)(uintptr_t)(
            src + ((size_t)b * FC + (size_t)c0) * (size_t)FHW + hw0);
        const unsigned lds_off = (unsigned)(uintptr_t)&tile[0][0];

        // ---- D# group 0: count=1 | lds_addr | global_addr | type=2 ----
        v4u g0;
        g0[0] = 1u;                                   // count=1 (valid)
        g0[1] = lds_off;                              // lds_addr (bytes)
        g0[2] = (unsigned)(ga & 0xFFFFFFFFu);         // global_addr[31:0]
        g0[3] = (unsigned)((ga >> 32) & 0x01FFFFFFu)  // global_addr[56:32]
              | (2u << 30);                           // type=2 ("image")

        // ---- D# group 1 ----
        // dword0: wg_mask=0 | data_size=2 (4B) @16 | pad_enable @20 |
        //         pad_interval=4 (32 dwords) @22 | pad_amount=0 (1 dword) @25
        v8i g1;
        g1[0] = (2 << 16) | (1 << 20) | (4 << 22);
        // dword1: atomic_barrier_addr[15:0]=0 | tensor_dim0[15:0] @48
        g1[1] = (int)(((unsigned)FHW & 0xFFFFu) << 16);
        // dword2: tensor_dim0[31:16] | tensor_dim1[15:0] @80
        g1[2] = (int)((((unsigned)FHW >> 16) & 0xFFFFu) |
                      (((unsigned)FC & 0xFFFFu) << 16));
        // dword3: tensor_dim1[31:16]=0 | tile_dim0=32 @112
        g1[3] = (int)(32u << 16);
        // dword4: tile_dim1=32 @128 | tile_dim2=0 @144
        g1[4] = 32;
        // dword5: tensor_dim0_stride[31:0]
        g1[5] = FHW;
        // dword6: tensor_dim0_stride[47:32]=0 | tensor_dim1_stride[15:0]=0
        g1[6] = 0;
        // dword7: tensor_dim1_stride[47:16]=0
        g1[7] = 0;

        v4i g2 = {0, 0, 0, 0};             // 2D tensor: groups 2/3 unused
        v4i g3 = {0, 0, 0, 0};
        v8i g4 = {0, 0, 0, 0, 0, 0, 0, 0}; // extra group (clang-23 6-arg form)

        __builtin_amdgcn_tensor_load_to_lds(g0, g1, g2, g3, g4, /*cpol=*/0);
        __builtin_amdgcn_s_wait_tensorcnt(0);
    }
#else
    {
        const int r = t >> 3;              // channel row inside tile
        const int q = (t & 7) << 2;        // hw col inside tile
        const float* g =
            src + ((size_t)b * FC + (size_t)(c0 + r)) * (size_t)FHW + hw0 + q;
#if USE_ASYNC_LDS
        #pragma unroll
        for (int k = 0; k < 4; ++k) {
            __builtin_amdgcn_global_load_async_to_lds_b32(
                (as1i_t)(uintptr_t)(g + k),
                (as3i_t)(unsigned)(uintptr_t)&tile[r][q + k],
                /*offset=*/0, /*cpol=*/0);
        }
        __builtin_amdgcn_s_wait_asynccnt(0);
#else
        float4 v = *(const float4*)g;
        tile[r][q + 0] = v.x;
        tile[r][q + 1] = v.y;
        tile[r][q + 2] = v.z;
        tile[r][q + 3] = v.w;
#endif
    }
#endif
    __syncthreads();

    // ---- store phase: lane covers out[b][hw0+i][c0+j4..j4+3] ----
    const int i  = t >> 3;             // hw inside tile
    const int j4 = (t & 7) << 2;       // channel inside tile
    float4 o;
    o.x = tile[j4 + 0][i];
    o.y = tile[j4 + 1][i];
    o.z = tile[j4 + 2][i];
    o.w = tile[j4 + 3][i];
    float* d =
        dst + ((size_t)b * FHW + (size_t)(hw0 + i)) * (size_t)FC + c0 + j4;
    *(float4*)d = o;
}

// =====================================================================
// Kernel 2: RoIAlign gather from NHWC.  One 256-thread block per ROI,
// lane = channel.  Per bin: 4 coalesced 1KB corner loads + bilinear fma,
// plus global_prefetch of the next bin's corner rows.  Results staged in
// LDS as [c][bin] (stride 49: conflict-free), then streamed out with
// GLOBAL_STORE_ASYNC_FROM_LDS in the output's (c, ph, pw) order (fully
// coalesced; no VGPR round trip).
// =====================================================================
__global__ void roi_align_nhwc(const float* __restrict__ T,
                               const float* __restrict__ rois,
                               const int* __restrict__ bidx,
                               float* __restrict__ out) {
    __shared__ float res[ROI_ELEMS];   // 50176 B

    const int n = blockIdx.x;
    const int c = threadIdx.x;

    const float x1 = rois[n * 4 + 0] * SPATIAL_SCALE - 0.5f;
    const float y1 = rois[n * 4 + 1] * SPATIAL_SCALE - 0.5f;
    const float x2 = rois[n * 4 + 2] * SPATIAL_SCALE - 0.5f;
    const float y2 = rois[n * 4 + 3] * SPATIAL_SCALE - 0.5f;
    const float bw = (x2 - x1) * (1.0f / OWP);
    const float bh = (y2 - y1) * (1.0f / OHP);

    const float* base = T + (size_t)bidx[n] * (size_t)FHW * (size_t)FC;

    for (int bin = 0; bin < NBINS; ++bin) {
        const int ph = bin / OWP;
        const int pw = bin - ph * OWP;
        const float x = x1 + ((float)pw + 0.5f) * bw;
        const float y = y1 + ((float)ph + 0.5f) * bh;

        const bool valid =
            (y >= -1.0f) && (y <= (float)FH) && (x >= -1.0f) && (x <= (float)FW);

        const float yc = fminf(fmaxf(y, 0.0f), (float)(FH - 1));
        const float xc = fminf(fmaxf(x, 0.0f), (float)(FW - 1));
        const int yl = (int)floorf(yc);
        const int xl = (int)floorf(xc);
        const int yh = (yl + 1 > FH - 1) ? (FH - 1) : (yl + 1);
        const int xh = (xl + 1 > FW - 1) ? (FW - 1) : (xl + 1);
        const float ly = yc - (float)yl;
        const float lx = xc - (float)xl;
        const float hy = 1.0f - ly;
        const float hx = 1.0f - lx;

        const float* r0 = base + (size_t)(yl * FW) * FC;
        const float* r1 = base + (size_t)(yh * FW) * FC;
        const float v00 = r0[(size_t)xl * FC + c];
        const float v01 = r0[(size_t)xh * FC + c];
        const float v10 = r1[(size_t)xl * FC + c];
        const float v11 = r1[(size_t)xh * FC + c];

        const float v = hy * (hx * v00 + lx * v01) + ly * (hx * v10 + lx * v11);
        res[c * NBINS + bin] = valid ? v : 0.0f;

        // Prefetch next bin's corner rows (global_prefetch_b8).
        if (bin + 1 < NBINS) {
            const int bn  = bin + 1;
            const int phn = bn / OWP;
            const int pwn = bn - phn * OWP;
            const float xn =
                fminf(fmaxf(x1 + ((float)pwn + 0.5f) * bw, 0.0f), (float)(FW - 1));
            const float yn =
                fminf(fmaxf(y1 + ((float)phn + 0.5f) * bh, 0.0f), (float)(FH - 1));
            const float* pf =
                base + (size_t)((int)yn * FW + (int)xn) * FC + c;
            __builtin_prefetch(pf, 0, 1);
            __builtin_prefetch(pf + (size_t)FW * FC, 0, 1);
        }
    }
    __syncthreads();

    // Writeback: out[n][c][ph][pw] = res[c*49 + bin], streamed linearly.
    const size_t ob = (size_t)n * ROI_ELEMS;
#if USE_ASYNC_ST
    #pragma unroll
    for (int k = 0; k < NBINS; ++k) {
        const int idx = k * FC + c;
        __builtin_amdgcn_global_store_async_from_lds_b32(
            (as1i_t)(uintptr_t)(out + ob + idx),
            (as3i_t)(unsigned)(uintptr_t)&res[idx],
            /*offset=*/0, /*cpol=*/0);
    }
    __builtin_amdgcn_s_wait_asynccnt(0);
#else
    #pragma unroll
    for (int k = 0; k < NBINS; ++k) {
        const int idx = k * FC + c;
        out[ob + idx] = res[idx];
    }
#endif
}

// =====================================================================
// Fallback: direct scattered gather from NCHW (used if ws too small).
// One thread per output element.
// =====================================================================
__global__ void roi_align_direct(const float* __restrict__ feat,
                                 const float* __restrict__ rois,
                                 const int* __restrict__ bidx,
                                 float* __restrict__ out, int total) {
    const int i = blockIdx.x * 256 + threadIdx.x;
    if (i >= total) return;

    const int pw = i % OWP;
    int tt = i / OWP;
    const int ph = tt % OHP;
    tt /= OHP;
    const int c = tt % FC;
    const int n = tt / FC;

    const float x1 = rois[n * 4 + 0] * SPATIAL_SCALE - 0.5f;
    const float y1 = rois[n * 4 + 1] * SPATIAL_SCALE - 0.5f;
    const float x2 = rois[n * 4 + 2] * SPATIAL_SCALE - 0.5f;
    const float y2 = rois[n * 4 + 3] * SPATIAL_SCALE - 0.5f;
    const float bw = (x2 - x1) * (1.0f / OWP);
    const float bh = (y2 - y1) * (1.0f / OHP);

    const float x = x1 + ((float)pw + 0.5f) * bw;
    const float y = y1 + ((float)ph + 0.5f) * bh;
    const bool valid =
        (y >= -1.0f) && (y <= (float)FH) && (x >= -1.0f) && (x <= (float)FW);

    const float yc = fminf(fmaxf(y, 0.0f), (float)(FH - 1));
    const float xc = fminf(fmaxf(x, 0.0f), (float)(FW - 1));
    const int yl = (int)floorf(yc);
    const int xl = (int)floorf(xc);
    const int yh = (yl + 1 > FH - 1) ? (FH - 1) : (yl + 1);
    const int xh = (xl + 1 > FW - 1) ? (FW - 1) : (xl + 1);
    const float ly = yc - (float)yl;
    const float lx = xc - (float)xl;
    const float hy = 1.0f - ly;
    const float hx = 1.0f - lx;

    const float* ch =
        feat + ((size_t)bidx[n] * FC + (size_t)c) * (size_t)FHW;
    const float v00 = ch[yl * FW + xl];
    const float v01 = ch[yl * FW + xh];
    const float v10 = ch[yh * FW + xl];
    const float v11 = ch[yh * FW + xh];

    const float v = hy * (hx * v00 + lx * v01) + ly * (hx * v10 + lx * v11);
    out[i] = valid ? v : 0.0f;
}

// =====================================================================
extern "C" void kernel_launch(void* const* d_in, const int* in_sizes, int n_in,
                              void* d_out, int out_size, void* d_ws, size_t ws_size,
                              hipStream_t stream) {
    const float* feat = (const float*)d_in[0];
    const float* rois = (const float*)d_in[1];
    const int*   bidx = (const int*)d_in[2];
    float*       out  = (float*)d_out;

    const size_t need = (size_t)FB * FHW * FC * sizeof(float);  // ~275 MB
    if (ws_size >= need) {
        float* T = (float*)d_ws;
        nchw_to_nhwc<<<dim3(FHW / 32, FC / 32, FB), 256, 0, stream>>>(feat, T);
        roi_align_nhwc<<<NROIS, FC, 0, stream>>>(T, rois, bidx, out);
    } else {
        const int total = NROIS * FC * NBINS;
        roi_align_direct<<<(total + 255) / 256, 256, 0, stream>>>(
            feat, rois, bidx, out, total);
    }
}